// AdaptiveQuantizer_57767310131509
// MI455X (gfx1250) — compile-verified
//
#include <hip/hip_runtime.h>
#include <stdint.h>

// AdaptiveQuantizer for MI455X (gfx1250).
// B=16, C=256, H=W=64  ->  65536 pixels, 256 channels, f32.
// Memory-bound: floor = (67MB read + 67MB write) / 23.3 TB/s ~= 5.8 us.
// Per block: async-DMA a 256px x 256ch (256KB) tile into LDS
// (CDNA5 GLOBAL_LOAD_ASYNC_TO_LDS, ASYNCcnt-tracked), then do the channel
// min/max reduction AND the quantize pass out of LDS. HBM sees each element
// exactly once in, once out. All geometry is compile-time so every address
// folds to base + immediate offset (no int div, no 64-bit muls).

#define TPB  256        // threads per block == pixels per tile
#define CCH  256        // channels
constexpr int HW  = 4096;            // H*W (64*64)
constexpr int LOG2HW = 12;

typedef int v4i __attribute__((ext_vector_type(4)));
typedef __attribute__((address_space(1))) v4i* gv4_ptr;   // global int4*
typedef __attribute__((address_space(3))) v4i* lv4_ptr;   // LDS int4*

__device__ __forceinline__ void async_b128_to_lds(const void* gsrc, void* ldst) {
#if __has_builtin(__builtin_amdgcn_global_load_async_to_lds_b128)
  __builtin_amdgcn_global_load_async_to_lds_b128(
      (gv4_ptr)(uintptr_t)gsrc,
      (lv4_ptr)(uint32_t)(uintptr_t)ldst,   // low 32 bits of generic = LDS offset
      0, 0);
#else
  uint32_t lds_off = (uint32_t)(uintptr_t)ldst;
  asm volatile("global_load_async_to_lds_b128 %0, %1, off"
               :: "v"(lds_off), "v"(gsrc) : "memory");
#endif
}

__device__ __forceinline__ void wait_async_zero() {
#if __has_builtin(__builtin_amdgcn_s_wait_asynccnt)
  __builtin_amdgcn_s_wait_asynccnt(0);
#else
  asm volatile("s_wait_asynccnt 0" ::: "memory");
#endif
}

__global__ __launch_bounds__(TPB)
void AdaptiveQuantizer_gfx1250_kernel(const float* __restrict__ feat,
                                      const int*  __restrict__ bit_alloc,
                                      float* __restrict__ out) {
  // tile[c][px] : 256 x 256 f32 = 256KB (needs CDNA5's 320KB WGP LDS)
  __shared__ float tile[CCH * TPB];

  const int t   = threadIdx.x;
  const int P0  = blockIdx.x * TPB;       // first global pixel of this tile
  const int b   = P0 >> LOG2HW;           // batch index (tiles never straddle batches)
  const int hw0 = P0 & (HW - 1);

  const float* gbase = feat + ((size_t)b << (LOG2HW + 8)) + hw0;  // + (c<<12) + px

  // ---- Stage tile via async DMA: 16384 x b128 chunks, 64 per thread-strip.
  // chunk idx -> channel c = idx>>6, in-row chunk j = idx&63 (64 chunks/1KB row).
  // Global rows are 1KB-aligned (hw0 % 256 == 0) so every chunk is 16B-aligned.
#pragma unroll
  for (int k = 0; k < (CCH * TPB) / (4 * TPB); ++k) {             // 64 iterations
    int idx = k * TPB + t;
    int c   = idx >> 6;
    int j   = idx & 63;
    async_b128_to_lds(gbase + (c << LOG2HW) + j * 4, &tile[idx * 4]);
  }
  wait_async_zero();      // this wave's async writes to LDS are complete
  __syncthreads();        // -> all waves' staging complete

  // ---- Pass A: per-pixel channel min/max from LDS (lane==px -> bank-conflict-free)
  float fmin = tile[t];
  float fmax = fmin;
#pragma unroll 16
  for (int c = 1; c < CCH; ++c) {
    float v = tile[c * TPB + t];
    fmin = fminf(fmin, v);
    fmax = fmaxf(fmax, v);
  }

  const float rng    = fmax - fmin;
  const bool  valid  = fabsf(rng) > 1e-8f;
  const float invRng = valid ? (1.0f / rng) : 1.0f;

  int nb = bit_alloc[(b << LOG2HW) + hw0 + t];
  nb = nb < 1 ? 1 : (nb > 8 ? 8 : nb);
  const float Lm1    = (float)((1 << nb) - 1);
  const float invLm1 = 1.0f / Lm1;

  // ---- Pass B: quantize/dequantize from LDS, stream results out (NT stores,
  // c*HW folds into the 24-bit immediate offset: max 255*16KB = 4.18MB < 8.3MB)
  float* obase = out + ((size_t)b << (LOG2HW + 8)) + hw0 + t;
#pragma unroll 8
  for (int c = 0; c < CCH; ++c) {
    float v    = tile[c * TPB + t];
    float norm = (v - fmin) * invRng;
    norm = norm < 0.0f ? 0.0f : (norm > 1.0f ? 1.0f : norm);
    float q = __builtin_rintf(norm * Lm1) * invLm1;   // v_rndne_f32: half-to-even
    float o = valid ? (q * rng + fmin) : v;
    // nan_to_num(nan->0, +inf->1, -inf->-1) — identity for finite inputs
    if (o != o)                     o = 0.0f;
    else if (o >  3.402823466e38f)  o = 1.0f;
    else if (o < -3.402823466e38f)  o = -1.0f;
    __builtin_nontemporal_store(o, obase + (c << LOG2HW));
  }
}

extern "C" void kernel_launch(void* const* d_in, const int* in_sizes, int n_in,
                              void* d_out, int out_size, void* d_ws, size_t ws_size,
                              hipStream_t stream) {
  const float* feat = (const float*)d_in[0];
  const int*   bits = (const int*)d_in[1];
  float*       out  = (float*)d_out;

  const int nPix   = in_sizes[1];      // B*H*W = 65536
  const int blocks = nPix / TPB;       // 256 workgroups, 1 per WGP (LDS-bound)

  hipLaunchKernelGGL(AdaptiveQuantizer_gfx1250_kernel,
                     dim3(blocks), dim3(TPB), 0, stream,
                     feat, bits, out);
}